// MaskRasterizationLoss_2705829396671
// MI455X (gfx1250) — compile-verified
//
#include <hip/hip_runtime.h>
#include <math.h>

typedef __attribute__((ext_vector_type(2))) float v2f;
typedef __attribute__((ext_vector_type(8))) float v8f;

#define N_MASKS 128
#define NV      128
#define IMG     512
#define RES     64
#define INV_S   0.1f
#define PIX     (RES * RES)   // 4096
#define BLK     256           // threads per raster block (8 wave32)
#define PB      (PIX / BLK)   // 16 pixel-blocks per mask

// ---------------------------------------------------------------------------
// Wave-level exact fp32 sum of nvals LDS floats (a multiple of 64) using V_WMMA_F32_16X16X4_F32.
// B = all ones  =>  D[m,n] = sum_k A[m,k] + C[m,n]  (exact f32 FMA chain).
// Each chained WMMA folds 64 values; lane supplies A as v2f:
//   A element (M,K): lanes 0-15 -> K={0,1}, lanes 16-31 -> K={2,3}; M = lane&15.
// Value index mapping idx = K*16 + M covers 0..63 bijectively, so the grand
// total is exact regardless of row assignment. Columns of D are identical;
// lanes 0-15 hold rows 0-7, lanes 16-31 rows 8-15 -> one shfl_xor(16) fixup.
// Must be executed by a full wave (EXEC all ones).
// ---------------------------------------------------------------------------
__device__ __forceinline__ float wave_wmma_sum(const float* lds, int nvals, int lane) {
    v2f ones; ones.x = 1.0f; ones.y = 1.0f;
    v8f c = {};
    const int base = ((lane >> 4) << 5) + (lane & 15);
    for (int chunk = 0; chunk < nvals; chunk += 64) {
        const float* p = lds + chunk;
        v2f a; a.x = p[base]; a.y = p[base + 16];
        c = __builtin_amdgcn_wmma_f32_16x16x4_f32(
                /*neg_a=*/false, a, /*neg_b=*/false, ones,
                /*c_mod=*/(short)0, c, /*reuse_a=*/false, /*reuse_b=*/false);
    }
    float s = c[0] + c[1] + c[2] + c[3] + c[4] + c[5] + c[6] + c[7];
    s += __shfl_xor(s, 16);   // combine the two half-wave row groups
    return s;                 // valid on all lanes
}

// ---------------------------------------------------------------------------
// Kernel A: union bbox per mask. 1 wave per mask.
// ---------------------------------------------------------------------------
__global__ void ub_kernel(const float* __restrict__ preds,
                          const float* __restrict__ bboxes,
                          float* __restrict__ ws_ub) {
    const int n = blockIdx.x;
    const int lane = threadIdx.x;   // 32 threads
    float mnx = 1e30f, mny = 1e30f, mxx = -1e30f, mxy = -1e30f;
    const float* p = preds + (size_t)n * NV * 2;
    for (int v = lane; v < NV; v += 32) {
        const float x = p[v * 2 + 0];
        const float y = p[v * 2 + 1];
        mnx = fminf(mnx, x); mny = fminf(mny, y);
        mxx = fmaxf(mxx, x); mxy = fmaxf(mxy, y);
    }
    for (int off = 16; off >= 1; off >>= 1) {
        mnx = fminf(mnx, __shfl_xor(mnx, off));
        mny = fminf(mny, __shfl_xor(mny, off));
        mxx = fmaxf(mxx, __shfl_xor(mxx, off));
        mxy = fmaxf(mxy, __shfl_xor(mxy, off));
    }
    if (lane == 0) {
        const float* bb = bboxes + n * 4;
        ws_ub[n * 4 + 0] = fminf(mnx, bb[0]);
        ws_ub[n * 4 + 1] = fminf(mny, bb[1]);
        ws_ub[n * 4 + 2] = fmaxf(mxx, bb[2]);
        ws_ub[n * 4 + 3] = fmaxf(mxy, bb[3]);
    }
}

// ---------------------------------------------------------------------------
// Kernel B: fused crop_and_resize(tgt) + soft rasterization + dice partials.
// grid = (PB, N_MASKS), block = 256. Deterministic per-block partial sums.
// ---------------------------------------------------------------------------
__global__ void raster_kernel(const float* __restrict__ preds,
                              const float* __restrict__ masks,
                              const float* __restrict__ ws_ub,
                              float* __restrict__ ws_part) {
    __shared__ float2 verts[NV];
    __shared__ float  red[3][BLK];

    const int n   = blockIdx.y;
    const int tid = threadIdx.x;

    const float4 ub = ((const float4*)ws_ub)[n];
    const float invw = 1.0f / (ub.z - ub.x);
    const float invh = 1.0f / (ub.w - ub.y);

    if (tid < NV) {
        const float vx0 = preds[((size_t)n * NV + tid) * 2 + 0];
        const float vy0 = preds[((size_t)n * NV + tid) * 2 + 1];
        float2 v;
        v.x = (vx0 - ub.x) * invw * (float)RES - 0.5f;
        v.y = (vy0 - ub.y) * invh * (float)RES - 0.5f;
        verts[tid] = v;
    }
    __syncthreads();

    const int   pix = blockIdx.x * BLK + tid;
    const int   ix  = pix & (RES - 1);
    const int   iy  = pix >> 6;
    const float pxf = (float)ix;
    const float pyf = (float)iy;

    // ---- target: bilinear sample of 512x512 mask, threshold 0.5 ----
    const float xs = ub.x + (ub.z - ub.x) * ((float)ix + 0.5f) * (1.0f / RES) - 0.5f;
    const float ys = ub.y + (ub.w - ub.y) * ((float)iy + 0.5f) * (1.0f / RES) - 0.5f;
    const float x0 = floorf(xs), y0 = floorf(ys);
    const float wx = xs - x0,    wy = ys - y0;
    int x0i = (int)x0; x0i = min(max(x0i, 0), IMG - 1);
    int y0i = (int)y0; y0i = min(max(y0i, 0), IMG - 1);
    const int x1i = min(x0i + 1, IMG - 1);
    const int y1i = min(y0i + 1, IMG - 1);
    const float* mk = masks + (size_t)n * IMG * IMG;
    const float m00 = mk[y0i * IMG + x0i];
    const float m01 = mk[y0i * IMG + x1i];
    const float m10 = mk[y1i * IMG + x0i];
    const float m11 = mk[y1i * IMG + x1i];
    const float samp = (1.0f - wy) * ((1.0f - wx) * m00 + wx * m01)
                     +         wy  * ((1.0f - wx) * m10 + wx * m11);
    const float tg = (samp >= 0.5f) ? 1.0f : 0.0f;

    // ---- soft rasterization: min point-segment d^2 + crossing parity ----
    float d2min = 1e30f;
    int crossings = 0;
    #pragma unroll 4
    for (int v = 0; v < NV; ++v) {
        const float2 a = verts[v];
        const float2 b = verts[(v + 1) & (NV - 1)];
        const float abx = b.x - a.x, aby = b.y - a.y;
        const float denom = abx * abx + aby * aby;
        const float pax = pxf - a.x, pay = pyf - a.y;
        float t = (pax * abx + pay * aby) / (denom + 1e-12f);
        t = fminf(fmaxf(t, 0.0f), 1.0f);
        const float dx  = pax - t * abx;
        const float dyv = pay - t * aby;
        d2min = fminf(d2min, dx * dx + dyv * dyv);
        const bool  cond = (a.y > pyf) != (b.y > pyf);
        const float sdy  = (aby == 0.0f) ? 1.0f : aby;
        const float xint = a.x + (pyf - a.y) * abx / sdy;
        crossings += (cond && (pxf < xint)) ? 1 : 0;
    }
    const float sgn = (crossings & 1) ? 1.0f : -1.0f;
    float m = 1.0f / (1.0f + expf(-(sgn * d2min * INV_S)));
    m = fminf(fmaxf(m, 1e-5f), 0.99999f);

    red[0][tid] = m * tg;   // intersection
    red[1][tid] = m;        // sum(pred)
    red[2][tid] = tg;       // sum(tgt)
    __syncthreads();

    if (tid < 32) {         // wave 0, full EXEC: WMMA tree reductions
        const float s0 = wave_wmma_sum(red[0], BLK, tid);
        const float s1 = wave_wmma_sum(red[1], BLK, tid);
        const float s2 = wave_wmma_sum(red[2], BLK, tid);
        if (tid == 0) {
            float* out = ws_part + ((size_t)n * PB + blockIdx.x) * 3;
            out[0] = s0; out[1] = s1; out[2] = s2;
        }
    }
}

// ---------------------------------------------------------------------------
// Kernel C: per-mask dice loss + WMMA-reduced mean.
// ---------------------------------------------------------------------------
__global__ void finalize_kernel(const float* __restrict__ ws_part,
                                float* __restrict__ out) {
    __shared__ float losses[N_MASKS];
    const int tid = threadIdx.x;    // 128 threads
    float inter = 0.0f, si = 0.0f, st = 0.0f;
    const float* p = ws_part + (size_t)tid * PB * 3;
    #pragma unroll
    for (int b = 0; b < PB; ++b) {  // deterministic serial fold
        inter += p[b * 3 + 0];
        si    += p[b * 3 + 1];
        st    += p[b * 3 + 2];
    }
    losses[tid] = 1.0f - (2.0f * inter + 1.0f) / (si + st + 1.0f);
    __syncthreads();
    if (tid < 32) {
        const float s = wave_wmma_sum(losses, N_MASKS, tid);
        if (tid == 0) out[0] = s * (1.0f / (float)N_MASKS);  // LOSS_WEIGHT = 1
    }
}

// ---------------------------------------------------------------------------
extern "C" void kernel_launch(void* const* d_in, const int* in_sizes, int n_in,
                              void* d_out, int out_size, void* d_ws, size_t ws_size,
                              hipStream_t stream) {
    const float* preds  = (const float*)d_in[0];   // (128,128,2)
    const float* masks  = (const float*)d_in[1];   // (128,512,512)
    const float* bboxes = (const float*)d_in[2];   // (128,4)

    float* ws      = (float*)d_ws;
    float* ws_ub   = ws;           // 128*4 floats
    float* ws_part = ws + 512;     // 128*16*3 floats

    ub_kernel      <<<dim3(N_MASKS),     dim3(32),  0, stream>>>(preds, bboxes, ws_ub);
    raster_kernel  <<<dim3(PB, N_MASKS), dim3(BLK), 0, stream>>>(preds, masks, ws_ub, ws_part);
    finalize_kernel<<<dim3(1),           dim3(N_MASKS), 0, stream>>>(ws_part, (float*)d_out);
}